// MalwareGAT_85495618994219
// MI455X (gfx1250) — compile-verified
//
#include <hip/hip_runtime.h>
#include <cmath>

typedef __attribute__((ext_vector_type(2))) float v2f;
typedef __attribute__((ext_vector_type(8))) float v8f;

#define GAT_SLOPE 0.2f

// ---------- helpers ----------
__device__ __forceinline__ unsigned f2key(float f) {
  unsigned u = __float_as_uint(f);
  return (u & 0x80000000u) ? ~u : (u | 0x80000000u);   // monotone map: float order -> uint order
}
__device__ __forceinline__ float key2f(unsigned k) {
  unsigned u = (k & 0x80000000u) ? (k & 0x7fffffffu) : ~k;
  return __uint_as_float(u);
}
// edge e in [0, E+N): first E come from edge_index, last N are self-loops
__device__ __forceinline__ void edge_sd(const int* __restrict__ ei, int E, int e, int& s, int& d) {
  if (e < E) { s = ei[e]; d = ei[E + e]; }
  else       { s = e - E; d = s; }
}

// ---------- fp32 WMMA GEMM: C[M,N] = A[M,K] @ B[K,N] + bias[N] ----------
// One wave computes one 16x16 tile using V_WMMA_F32_16X16X4_F32 (exact fp32).
// M % 16 == 0, N % 16 == 0, K % 4 == 0 (holds for all GEMMs here).
__global__ void gemm16_wmma_f32(const float* __restrict__ A, const float* __restrict__ B,
                                const float* __restrict__ bias, float* __restrict__ C,
                                int M, int N, int K) {
  const int wave = (int)((blockIdx.x * blockDim.x + threadIdx.x) >> 5);
  const int lane = (int)(threadIdx.x & 31);
  const int ntiles = N >> 4;
  const int mt = wave / ntiles;
  const int nt = wave - mt * ntiles;
  if (mt * 16 >= M) return;                      // uniform per wave

  const int r     = lane & 15;                   // row (A) / col (B,D) within tile
  const int khalf = (lane >> 4) << 1;            // lanes 0-15: K {0,1}; lanes 16-31: K {2,3}

  const float* __restrict__ arow = A + (size_t)(mt * 16 + r) * K + khalf;
  const float* __restrict__ bcol = B + (size_t)khalf * N + (nt * 16 + r);

  v8f acc = {};
  for (int k = 0; k < K; k += 4) {
    v2f a; a.x = arow[0]; a.y = arow[1];
    v2f b; b.x = bcol[0]; b.y = bcol[N];
    acc = __builtin_amdgcn_wmma_f32_16x16x4_f32(false, a, false, b, (short)0, acc, false, false);
    arow += 4;
    bcol += (size_t)4 * N;
  }

  const int col   = nt * 16 + r;
  const float bv  = bias ? bias[col] : 0.0f;
  const int rbase = mt * 16 + ((lane >> 4) << 3); // lanes 0-15: rows 0-7; 16-31: rows 8-15
#pragma unroll
  for (int v = 0; v < 8; ++v)
    C[(size_t)(rbase + v) * N + col] = acc[v] + bv;
}

// ---------- per-node attention scores: s_src/s_dst [n, heads] ----------
__global__ void attn_scores(const float* __restrict__ h, const float* __restrict__ att_s,
                            const float* __restrict__ att_d, float* __restrict__ s_src,
                            float* __restrict__ s_dst, int n, int heads, int ch) {
  const int i = blockIdx.x * blockDim.x + threadIdx.x;     // node*heads + head
  if (i >= n * heads) return;
  const int node = i / heads, hd = i - node * heads;
  const float* __restrict__ hp = h + (size_t)node * heads * ch + (size_t)hd * ch;
  const float* __restrict__ as = att_s + hd * ch;
  const float* __restrict__ ad = att_d + hd * ch;
  float ss = 0.f, sd = 0.f;
  for (int c = 0; c < ch; ++c) { const float v = hp[c]; ss += v * as[c]; sd += v * ad[c]; }
  s_src[i] = ss;
  s_dst[i] = sd;
}

// ---------- segment max over edges (float via monotone uint keys) ----------
__global__ void edge_max(const int* __restrict__ ei, int E, int Nn,
                         const float* __restrict__ s_src, const float* __restrict__ s_dst,
                         unsigned* __restrict__ mkey, int heads) {
  const long i = (long)blockIdx.x * blockDim.x + threadIdx.x;
  const long tot = (long)(E + Nn) * heads;
  if (i >= tot) return;
  const int e = (int)(i / heads), hd = (int)(i - (long)e * heads);
  int s, d; edge_sd(ei, E, e, s, d);
  float x = s_src[(size_t)s * heads + hd] + s_dst[(size_t)d * heads + hd];
  x = x > 0.f ? x : GAT_SLOPE * x;
  atomicMax(&mkey[(size_t)d * heads + hd], f2key(x));
}

// ---------- segment sum of exp(e - m) ----------
__global__ void edge_denom(const int* __restrict__ ei, int E, int Nn,
                           const float* __restrict__ s_src, const float* __restrict__ s_dst,
                           const unsigned* __restrict__ mkey, float* __restrict__ den, int heads) {
  const long i = (long)blockIdx.x * blockDim.x + threadIdx.x;
  const long tot = (long)(E + Nn) * heads;
  if (i >= tot) return;
  const int e = (int)(i / heads), hd = (int)(i - (long)e * heads);
  int s, d; edge_sd(ei, E, e, s, d);
  float x = s_src[(size_t)s * heads + hd] + s_dst[(size_t)d * heads + hd];
  x = x > 0.f ? x : GAT_SLOPE * x;
  float m = key2f(mkey[(size_t)d * heads + hd]);
  if (!isfinite(m)) m = 0.f;
  atomicAdd(&den[(size_t)d * heads + hd], expf(x - m));
}

// ---------- alpha-weighted scatter-add; one wave per edge, lanes stripe features ----------
template <int HEADS, int CH>
__global__ void edge_aggregate(const int* __restrict__ ei, int E, int Nn,
                               const float* __restrict__ s_src, const float* __restrict__ s_dst,
                               const unsigned* __restrict__ mkey, const float* __restrict__ den,
                               const float* __restrict__ h, float* __restrict__ out) {
  constexpr int F = HEADS * CH;
  const int wave = (int)((blockIdx.x * blockDim.x + threadIdx.x) >> 5);
  const int lane = (int)(threadIdx.x & 31);
  if (wave >= E + Nn) return;
  int s, d; edge_sd(ei, E, wave, s, d);

  float alpha[HEADS];
#pragma unroll
  for (int hd = 0; hd < HEADS; ++hd) {
    float x = s_src[(size_t)s * HEADS + hd] + s_dst[(size_t)d * HEADS + hd];
    x = x > 0.f ? x : GAT_SLOPE * x;
    float m = key2f(mkey[(size_t)d * HEADS + hd]);
    if (!isfinite(m)) m = 0.f;
    alpha[hd] = expf(x - m) / (den[(size_t)d * HEADS + hd] + 1e-16f);
  }

  const float* __restrict__ hs = h + (size_t)s * F;
  float* __restrict__ od = out + (size_t)d * F;
#pragma unroll
  for (int j = 0; j < F / 32; ++j) {
    const int f = lane + 32 * j;                  // coalesced 128B per step
    atomicAdd(&od[f], hs[f] * alpha[(32 * j) / CH]);  // head index is compile-time per j
  }
}

// ---------- elementwise ELU ----------
__global__ void elu_kernel(float* __restrict__ x, long n) {
  const long i = (long)blockIdx.x * blockDim.x + threadIdx.x;
  if (i >= n) return;
  const float v = x[i];
  x[i] = v > 0.f ? v : expm1f(v);
}

// ---------- global mean pool (sums + counts) ----------
__global__ void pool_sum(const float* __restrict__ h, const int* __restrict__ batch,
                         float* __restrict__ sums, int n) {
  const long i = (long)blockIdx.x * blockDim.x + threadIdx.x;
  if (i >= (long)n * 64) return;
  const int node = (int)(i >> 6), f = (int)(i & 63);
  atomicAdd(&sums[(size_t)batch[node] * 64 + f], h[i]);
}
__global__ void pool_cnt(const int* __restrict__ batch, float* __restrict__ cnt, int n) {
  const int i = blockIdx.x * blockDim.x + threadIdx.x;
  if (i >= n) return;
  atomicAdd(&cnt[batch[i]], 1.0f);
}

// ---------- pooled @ fc + log_softmax (64 graphs, 2 classes) ----------
__global__ void head_kernel(const float* __restrict__ sums, const float* __restrict__ cnt,
                            const float* __restrict__ fcw, const float* __restrict__ fcb,
                            float* __restrict__ out) {
  const int g = threadIdx.x;
  if (g >= 64) return;
  float c = cnt[g]; c = c > 1.f ? c : 1.f;
  float l0 = fcb[0], l1 = fcb[1];
  for (int k = 0; k < 64; ++k) {
    const float p = sums[(size_t)g * 64 + k] / c;
    l0 += p * fcw[k * 2 + 0];
    l1 += p * fcw[k * 2 + 1];
  }
  const float mx  = fmaxf(l0, l1);
  const float lse = mx + logf(expf(l0 - mx) + expf(l1 - mx));
  out[g * 2 + 0] = l0 - lse;
  out[g * 2 + 1] = l1 - lse;
}

static inline int cdiv(long a, long b) { return (int)((a + b - 1) / b); }

extern "C" void kernel_launch(void* const* d_in, const int* in_sizes, int n_in,
                              void* d_out, int out_size, void* d_ws, size_t ws_size,
                              hipStream_t stream) {
  const float* x    = (const float*)d_in[0];
  const int*   ei   = (const int*)d_in[1];
  const int*   batch= (const int*)d_in[2];
  const float* W1   = (const float*)d_in[3];
  const float* as1  = (const float*)d_in[4];
  const float* ad1  = (const float*)d_in[5];
  const float* b1   = (const float*)d_in[6];
  const float* W2   = (const float*)d_in[7];
  const float* as2  = (const float*)d_in[8];
  const float* ad2  = (const float*)d_in[9];
  const float* b2   = (const float*)d_in[10];
  const float* fcw  = (const float*)d_in[11];
  const float* fcb  = (const float*)d_in[12];
  float* out = (float*)d_out;

  const int Nn   = in_sizes[0] / 128;   // 50000
  const int E    = in_sizes[1] / 2;     // 800000
  const int Etot = E + Nn;
  const int F1 = 256, F2 = 64, Kin = 128;

  // Workspace layout (floats). h2/agg2 reuse the h1 region once layer-1 agg is done.
  float* ws   = (float*)d_ws;
  float* h1   = ws;                        // Nn*256
  float* h2   = ws;                        // Nn*64   (overlays h1, used after h1 dead)
  float* agg2 = ws + (size_t)Nn * F2;      // Nn*64   (overlays h1 tail)
  float* agg1 = ws + (size_t)Nn * F1;      // Nn*256  (becomes x2 after ELU)
  float* p    = ws + (size_t)2 * Nn * F1;
  float*    ssrc1 = p;              p += (size_t)Nn * 4;
  float*    sdst1 = p;              p += (size_t)Nn * 4;
  unsigned* mkey1 = (unsigned*)p;   p += (size_t)Nn * 4;
  float*    den1  = p;              p += (size_t)Nn * 4;
  float*    ssrc2 = p;              p += Nn;
  float*    sdst2 = p;              p += Nn;
  unsigned* mkey2 = (unsigned*)p;   p += Nn;
  float*    den2  = p;              p += Nn;
  float*    sums  = p;              p += 64 * 64;
  float*    cnt   = p;              p += 64;
  // total ~= (2*Nn*256 + Nn*20 + 4160) floats ~= 106.5 MB

  // zero-init (key 0 decodes to NaN -> reference's isfinite->0 path)
  hipMemsetAsync(agg1, 0, (size_t)Nn * F1 * sizeof(float), stream);
  hipMemsetAsync(mkey1, 0, (size_t)Nn * 4 * sizeof(unsigned), stream);
  hipMemsetAsync(den1, 0, (size_t)Nn * 4 * sizeof(float), stream);
  hipMemsetAsync(mkey2, 0, (size_t)Nn * sizeof(unsigned), stream);
  hipMemsetAsync(den2, 0, (size_t)Nn * sizeof(float), stream);
  hipMemsetAsync(sums, 0, 64 * 64 * sizeof(float), stream);
  hipMemsetAsync(cnt, 0, 64 * sizeof(float), stream);

  // ---- layer 1 ----
  const long waves1 = (long)(Nn / 16) * (F1 / 16);
  gemm16_wmma_f32<<<cdiv(waves1 * 32, 256), 256, 0, stream>>>(x, W1, b1, h1, Nn, F1, Kin);
  attn_scores<<<cdiv((long)Nn * 4, 256), 256, 0, stream>>>(h1, as1, ad1, ssrc1, sdst1, Nn, 4, 64);
  edge_max<<<cdiv((long)Etot * 4, 256), 256, 0, stream>>>(ei, E, Nn, ssrc1, sdst1, mkey1, 4);
  edge_denom<<<cdiv((long)Etot * 4, 256), 256, 0, stream>>>(ei, E, Nn, ssrc1, sdst1, mkey1, den1, 4);
  edge_aggregate<4, 64><<<cdiv((long)Etot * 32, 256), 256, 0, stream>>>(
      ei, E, Nn, ssrc1, sdst1, mkey1, den1, h1, agg1);
  elu_kernel<<<cdiv((long)Nn * F1, 256), 256, 0, stream>>>(agg1, (long)Nn * F1);

  // ---- layer 2 (h1 region now dead; h2/agg2 overlay it) ----
  const long waves2 = (long)(Nn / 16) * (F2 / 16);
  gemm16_wmma_f32<<<cdiv(waves2 * 32, 256), 256, 0, stream>>>(agg1, W2, b2, h2, Nn, F2, F1);
  attn_scores<<<cdiv((long)Nn, 256), 256, 0, stream>>>(h2, as2, ad2, ssrc2, sdst2, Nn, 1, 64);
  hipMemsetAsync(agg2, 0, (size_t)Nn * F2 * sizeof(float), stream);
  edge_max<<<cdiv((long)Etot, 256), 256, 0, stream>>>(ei, E, Nn, ssrc2, sdst2, mkey2, 1);
  edge_denom<<<cdiv((long)Etot, 256), 256, 0, stream>>>(ei, E, Nn, ssrc2, sdst2, mkey2, den2, 1);
  edge_aggregate<1, 64><<<cdiv((long)Etot * 32, 256), 256, 0, stream>>>(
      ei, E, Nn, ssrc2, sdst2, mkey2, den2, h2, agg2);
  elu_kernel<<<cdiv((long)Nn * F2, 256), 256, 0, stream>>>(agg2, (long)Nn * F2);

  // ---- pool + FC + log_softmax ----
  pool_sum<<<cdiv((long)Nn * 64, 256), 256, 0, stream>>>(agg2, batch, sums, Nn);
  pool_cnt<<<cdiv((long)Nn, 256), 256, 0, stream>>>(batch, cnt, Nn);
  head_kernel<<<1, 64, 0, stream>>>(sums, cnt, fcw, fcb, out);
}